// AliasFreeConv_80427557584919
// MI455X (gfx1250) — compile-verified
//
#include <hip/hip_runtime.h>
#include <hip/hip_bf16.h>

// ---------------- problem constants (fixed by reference) ----------------
#define BATCH 8
#define CI    512
#define CO    512
#define HIN   64
#define SDIM  512
#define HC    66            // conv output spatial (64 + 3 - 1)
#define NPIX  (HC*HC)       // 4356
#define TAPS  12
#define NCHUNK (9 * (CI / 16))   // 288 K-chunks (9 taps x 32 ci-chunks)
#define WSTR  20            // W tile LDS row stride (80B: b128-aligned rows)
#define XSTR  18            // X tile LDS row stride (72B: b64-aligned pairs)

typedef float v2f __attribute__((ext_vector_type(2)));
typedef float v8f __attribute__((ext_vector_type(8)));

// ---- CDNA5 async global->LDS (ASYNCcnt-tracked), inline asm per ISA 15.18.3
__device__ __forceinline__ void async_ld_b128(const float* g, const float* l) {
    unsigned lds = (unsigned)(unsigned long long)l;     // low 32b = LDS offset
    unsigned long long ga = (unsigned long long)g;
    asm volatile("global_load_async_to_lds_b128 %0, %1, off"
                 :: "v"(lds), "v"(ga) : "memory");
}
__device__ __forceinline__ void wait_async0() {
#if __has_builtin(__builtin_amdgcn_s_wait_asynccnt)
    __builtin_amdgcn_s_wait_asynccnt(0);
#else
    asm volatile("s_wait_asynccnt 0x0" ::: "memory");
#endif
}

// ---------------- kernel 1: s[b,ci] = style @ (mod_w^T / sqrt(S)) + mod_bias
__global__ __launch_bounds__(256) void modstyle_kernel(
    const float* __restrict__ style, const float* __restrict__ mw,
    const float* __restrict__ mb, float* __restrict__ s)
{
    int idx = blockIdx.x * 256 + threadIdx.x;          // b*CI + ci
    if (idx >= BATCH * CI) return;
    int b = idx >> 9, ci = idx & (CI - 1);
    const float scale = 0.044194173824159216f;          // 1/sqrt(512)
    const float* st = style + b * SDIM;
    const float* w  = mw + (size_t)ci * SDIM;
    float a = 0.f;
    for (int k = 0; k < SDIM; ++k) a += st[k] * w[k];
    s[idx] = a * scale + mb[ci];
}

// ---------------- kernel 2: wsq[co,ci] = sum_k weight^2
__global__ __launch_bounds__(256) void wsq_kernel(
    const float* __restrict__ weight, float* __restrict__ wsq)
{
    int idx = blockIdx.x * 256 + threadIdx.x;          // co*CI + ci
    if (idx >= CO * CI) return;
    const float* w = weight + (size_t)idx * 9;
    float a = 0.f;
#pragma unroll
    for (int t = 0; t < 9; ++t) a += w[t] * w[t];
    wsq[idx] = a;
}

// ---------------- kernel 3: dcoef[b,co] = rsqrt(sum_ci wsq*s^2 + eps) * rsqrt(ema)
__global__ __launch_bounds__(256) void dcoef_kernel(
    const float* __restrict__ wsq, const float* __restrict__ s,
    const float* __restrict__ ema, float* __restrict__ dcoef)
{
    int idx = blockIdx.x * 256 + threadIdx.x;          // b*CO + co
    if (idx >= BATCH * CO) return;
    int b = idx >> 9, co = idx & (CO - 1);
    const float* wr = wsq + (size_t)co * CI;
    const float* sr = s + (size_t)b * CI;
    float a = 0.f;
    for (int ci = 0; ci < CI; ++ci) { float sv = sr[ci]; a += wr[ci] * sv * sv; }
    dcoef[idx] = rsqrtf(a + 1e-8f) * rsqrtf(ema[0]);
}

// ---------------- kernel 4: xs = input * s[b,ci]  (fold modulation into input)
__global__ __launch_bounds__(256) void scale_input_kernel(
    const float* __restrict__ input, const float* __restrict__ s,
    float* __restrict__ xs)
{
    long long i = ((long long)blockIdx.x * 256 + threadIdx.x) * 4;
    if (i >= (long long)BATCH * CI * HIN * HIN) return;
    int b  = (int)(i >> 21);
    int ci = (int)((i >> 12) & (CI - 1));
    float sv = s[b * CI + ci];
    float4 v = *reinterpret_cast<const float4*>(input + i);
    v.x *= sv; v.y *= sv; v.z *= sv; v.w *= sv;
    *reinterpret_cast<float4*>(xs + i) = v;
}

// ---------------- kernel 5: weight -> wT[tap][co][ci] (ci contiguous for DMA)
__global__ __launch_bounds__(256) void wrearr_kernel(
    const float* __restrict__ weight, float* __restrict__ wT)
{
    int idx = blockIdx.x * 256 + threadIdx.x;          // over CO*CI*9
    if (idx >= CO * CI * 9) return;
    int tap = idx % 9;
    int rem = idx / 9;
    int ci = rem & (CI - 1);
    int co = rem >> 9;
    wT[((size_t)tap * CO + co) * CI + ci] = weight[idx];
}

// ---------------- kernel 6: implicit-GEMM 3x3 conv via V_WMMA_F32_16X16X4_F32
// Block = 8 waves; tile = 64 co x 64 px; double-buffered LDS; weights arrive
// by async global->LDS b128 DMA (ASYNCcnt), inputs by gathered im2col stores.
__global__ __launch_bounds__(256) void conv_wmma_kernel(
    const float* __restrict__ xs,      // (B,CI,64,64), pre-scaled by s
    const float* __restrict__ wT,      // (9,CO,CI)
    const float* __restrict__ dcoef,   // (B,CO)
    const float* __restrict__ bias,    // (CO)
    float* __restrict__ convout)       // (B,CO,66,66)
{
    __shared__ __align__(16) float Wlds[2][64][WSTR];
    __shared__ __align__(16) float Xlds[2][64][XSTR];

    const int tid  = threadIdx.x;
    const int lane = tid & 31;
    const int wave = tid >> 5;
    const int cb = (wave & 3) * 16;    // co sub-tile base within 64
    const int nb = (wave >> 2) * 32;   // pixel pair-subtile base within 64
    const int pbase  = blockIdx.x * 64;
    const int cobase = blockIdx.y * 64;
    const int b = blockIdx.z;

    const int m    = lane & 15;
    const int koff = (lane >> 4) << 1; // lanes 0-15: K{0,1}; 16-31: K{2,3}

    v8f acc0 = {}, acc1 = {};

    auto stage = [&](int c, int buf) {
        const int tap = c >> 5;             // 0..8
        const int cib = (c & 31) << 4;      // ci chunk base
        // ---- weights: one async b128 per lane covers the whole 64x16 tile
        {
            const int co = tid >> 2, seg = tid & 3;
            const float* g = wT + (((size_t)tap * CO + cobase + co) * CI + cib + seg * 4);
            async_ld_b128(g, &Wlds[buf][co][seg * 4]);
        }
        // ---- input im2col gather (zero-filled borders), pixel-major in LDS
        const int kh = tap / 3, kw = tap - kh * 3;
#pragma unroll
        for (int t = 0; t < 4; ++t) {
            int idx = tid + t * 256;
            int j = idx & 63, kci = idx >> 6;
            int p = pbase + j;
            int y = p / HC, x = p - y * HC;
            int iy = y + kh - 2, ix = x + kw - 2;
            float v = 0.f;
            if (p < NPIX && (unsigned)iy < (unsigned)HIN && (unsigned)ix < (unsigned)HIN)
                v = xs[(((size_t)b * CI + cib + kci) * HIN + iy) * HIN + ix];
            Xlds[buf][j][kci] = v;
        }
    };

    stage(0, 0);
    for (int c = 0; c < NCHUNK; ++c) {
        const int buf = c & 1;
        wait_async0();                 // my async LDS writes for chunk c done
        __syncthreads();               // everyone's staging published
        if (c + 1 < NCHUNK) stage(c + 1, buf ^ 1);   // overlap with WMMA below
#pragma unroll
        for (int kq = 0; kq < 16; kq += 4) {
            v2f a, b0, b1;
            a.x  = Wlds[buf][cb + m][kq + koff];
            a.y  = Wlds[buf][cb + m][kq + koff + 1];
            b0.x = Xlds[buf][nb + m][kq + koff];
            b0.y = Xlds[buf][nb + m][kq + koff + 1];
            b1.x = Xlds[buf][nb + 16 + m][kq + koff];
            b1.y = Xlds[buf][nb + 16 + m][kq + koff + 1];
            acc0 = __builtin_amdgcn_wmma_f32_16x16x4_f32(
                false, a, false, b0, (short)0, acc0, false, false);
            acc1 = __builtin_amdgcn_wmma_f32_16x16x4_f32(
                false, a, false, b1, (short)0, acc1, false, false);
        }
    }

    // ---- epilogue: row scale by demod*rsqrt(ema), add bias, store
    const int half = lane >> 4;
    const int n = lane & 15;
    const int p0 = pbase + nb + n;
    const int p1 = p0 + 16;
#pragma unroll
    for (int r = 0; r < 8; ++r) {
        int co = cobase + cb + r + half * 8;
        float d  = dcoef[b * CO + co];
        float bv = bias[co];
        size_t base = ((size_t)b * CO + co) * NPIX;
        if (p0 < NPIX) convout[base + p0] = acc0[r] * d + bv;
        if (p1 < NPIX) convout[base + p1] = acc1[r] * d + bv;
    }
}

// ---------------- kernel 7: filtered_lrelu, fully fused in LDS per channel
__global__ __launch_bounds__(256) void flrelu_kernel(
    const float* __restrict__ convout, const float* __restrict__ fup,
    const float* __restrict__ fdn, float* __restrict__ out)
{
    extern __shared__ __align__(16) float smem[];
    float* bufA = smem;                  // 4356 floats
    float* t1   = smem + 4356;           // 138*66 = 9108
    float* t2   = smem + 4356 + 9108;    // 138*138 = 19044
    float* t3   = smem;                  // 64*138 = 8832, aliases dead bufA+t1
    __shared__ float fu[TAPS], fd[TAPS];

    const int tid = threadIdx.x;
    const int ch  = blockIdx.x;          // b*CO + co
    if (tid < TAPS) { fu[tid] = fup[tid]; fd[tid] = fdn[tid]; }

    // async DMA the 66x66 tile into LDS: 4356 = 1089 x b128, 16B-aligned
    const float* src = convout + (size_t)ch * NPIX;
    for (int idx = tid; idx < NPIX / 4; idx += 256)
        async_ld_b128(src + idx * 4, bufA + idx * 4);
    wait_async0();
    __syncthreads();

    // up-FIR along H (zero-insertion up=2, pad 9/9): t1[i][x], i in 0..137
    for (int e = tid; e < 138 * 66; e += 256) {
        int i = e / 66, x = e - i * 66;
        float a = 0.f;
#pragma unroll
        for (int t = 0; t < TAPS; ++t) {
            int mm = i + t - 9;
            if (mm >= 0 && mm <= 130 && !(mm & 1))
                a += fu[t] * bufA[(mm >> 1) * 66 + x];
        }
        t1[e] = a;
    }
    __syncthreads();

    // up-FIR along W, then gain up^2=4 and lrelu * sqrt(2): t2[i][j]
    for (int e = tid; e < 138 * 138; e += 256) {
        int i = e / 138, j = e - i * 138;
        float a = 0.f;
#pragma unroll
        for (int t = 0; t < TAPS; ++t) {
            int mm = j + t - 9;
            if (mm >= 0 && mm <= 130 && !(mm & 1))
                a += fu[t] * t1[i * 66 + (mm >> 1)];
        }
        a *= 4.0f;
        a = (a >= 0.f ? a : 0.2f * a) * 1.4142135623730951f;
        t2[e] = a;
    }
    __syncthreads();

    // down-FIR along H, stride 2: t3[i][j], i in 0..63
    for (int e = tid; e < 64 * 138; e += 256) {
        int i = e / 138, j = e - i * 138;
        float a = 0.f;
#pragma unroll
        for (int t = 0; t < TAPS; ++t)
            a += fd[t] * t2[(2 * i + t) * 138 + j];
        t3[e] = a;
    }
    __syncthreads();

    // down-FIR along W, stride 2: out[y][x]
    float* dst = out + (size_t)ch * (HIN * HIN);
    for (int e = tid; e < HIN * HIN; e += 256) {
        int y = e >> 6, x = e & 63;
        float a = 0.f;
#pragma unroll
        for (int t = 0; t < TAPS; ++t)
            a += fd[t] * t3[y * 138 + 2 * x + t];
        dst[e] = a;
    }
}

// ---------------- host-side launcher ----------------
extern "C" void kernel_launch(void* const* d_in, const int* in_sizes, int n_in,
                              void* d_out, int out_size, void* d_ws, size_t ws_size,
                              hipStream_t stream) {
    (void)in_sizes; (void)n_in; (void)out_size; (void)ws_size;
    const float* input      = (const float*)d_in[0];
    const float* style      = (const float*)d_in[1];
    const float* weight     = (const float*)d_in[2];
    const float* mod_weight = (const float*)d_in[3];
    const float* mod_bias   = (const float*)d_in[4];
    const float* bias       = (const float*)d_in[5];
    const float* up_filter  = (const float*)d_in[6];
    const float* down_filter= (const float*)d_in[7];
    const float* ema_var    = (const float*)d_in[8];
    float* out = (float*)d_out;

    float* ws      = (float*)d_ws;
    float* s       = ws;                                    // 4096
    float* wsq     = s + BATCH * CI;                        // 262144
    float* dcoef   = wsq + (size_t)CO * CI;                 // 4096
    float* xs      = dcoef + BATCH * CO;                    // 16,777,216
    float* convout = xs + (size_t)BATCH * CI * HIN * HIN;   // 17,842,176
    float* wT      = convout + (size_t)BATCH * CO * NPIX;   // 2,359,296

    modstyle_kernel<<<(BATCH * CI + 255) / 256, 256, 0, stream>>>(style, mod_weight, mod_bias, s);
    wsq_kernel<<<(CO * CI + 255) / 256, 256, 0, stream>>>(weight, wsq);
    dcoef_kernel<<<(BATCH * CO + 255) / 256, 256, 0, stream>>>(wsq, s, ema_var, dcoef);
    scale_input_kernel<<<(BATCH * CI * HIN * HIN / 4 + 255) / 256, 256, 0, stream>>>(input, s, xs);
    wrearr_kernel<<<(CO * CI * 9 + 255) / 256, 256, 0, stream>>>(weight, wT);

    dim3 cgrid((NPIX + 63) / 64, CO / 64, BATCH);           // (69, 8, 8)
    conv_wmma_kernel<<<cgrid, 256, 0, stream>>>(xs, wT, dcoef, bias, convout);

    const int smem_bytes = (4356 + 9108 + 19044) * 4;       // 130,032 B dynamic LDS
    flrelu_kernel<<<BATCH * CO, 256, smem_bytes, stream>>>(convout, up_filter, down_filter, out);
}